// GNN_85349590106382
// MI455X (gfx1250) — compile-verified
//
#include <hip/hip_runtime.h>
#include <math.h>

typedef float v2f __attribute__((ext_vector_type(2)));
typedef float v8f __attribute__((ext_vector_type(8)));

#define DD 128

// ---------- helpers ----------
__device__ __forceinline__ float hswish(float x) {
    return x * fminf(fmaxf(x + 3.0f, 0.0f), 6.0f) * (1.0f / 6.0f);
}
__device__ __forceinline__ unsigned fflip(float f) {
    unsigned u = __float_as_uint(f);
    return (u & 0x80000000u) ? ~u : (u | 0x80000000u);
}
__device__ __forceinline__ float funflip(unsigned u) {
    return (u & 0x80000000u) ? __uint_as_float(u & 0x7fffffffu) : __uint_as_float(~u);
}

// ---------- weight fusion: Wf = (Wt @ W1) * bn_scale (per out col), bf folded ----------
__global__ void gnn_fuse_weights(const float* __restrict__ Wt, const float* __restrict__ bt,
                                 const float* __restrict__ W1, const float* __restrict__ b1,
                                 const float* __restrict__ gma, const float* __restrict__ bta,
                                 const float* __restrict__ mu,  const float* __restrict__ var,
                                 float* __restrict__ Wf, float* __restrict__ bf, int Lt) {
    int tid = blockIdx.x * blockDim.x + threadIdx.x;
    int total = Lt * DD * DD;
    if (tid >= total) return;
    int l = tid / (DD * DD);
    int rem = tid - l * DD * DD;
    int k = rem / DD, j = rem - (rem / DD) * DD;
    const float* wt = Wt + (size_t)l * DD * DD + (size_t)k * DD;
    const float* w1 = W1 + (size_t)l * DD * DD + j;
    float s = 0.0f;
    for (int t = 0; t < DD; ++t) s += wt[t] * w1[(size_t)t * DD];
    float scale = gma[l * DD + j] * rsqrtf(var[l * DD + j] + 1e-5f);
    Wf[tid] = s * scale;
    if (k == 0) {
        const float* btl = bt + l * DD;
        float sb = 0.0f;
        for (int t = 0; t < DD; ++t) sb += btl[t] * w1[(size_t)t * DD];
        bf[l * DD + j] = (sb + b1[l * DD + j] - mu[l * DD + j]) * scale + bta[l * DD + j];
    }
}

// ---------- per-node: gate scalar, init max (with self-loop), zero denom/agg ----------
__global__ void gnn_gate_init(const float* __restrict__ h, const float* __restrict__ Wg,
                              const float* __restrict__ bg, float* __restrict__ g,
                              unsigned* __restrict__ menc, float* __restrict__ denom,
                              float* __restrict__ aggX, int n) {
    int v = blockIdx.x * blockDim.x + threadIdx.x;
    if (v >= n) return;
    const float4* hr = (const float4*)(h + (size_t)v * DD);
    const float4* wg = (const float4*)Wg;
    float s = 0.0f;
#pragma unroll
    for (int i = 0; i < DD / 4; ++i) {
        float4 a = hr[i], b = wg[i];
        s += a.x * b.x + a.y * b.y + a.z * b.z + a.w * b.w;
    }
    s += bg[0];
    g[v] = s;
    menc[v] = fflip(s);      // self-loop participates in segment max
    denom[v] = 0.0f;
    float4* ar = (float4*)(aggX + (size_t)v * DD);
    float4 z = make_float4(0.f, 0.f, 0.f, 0.f);
#pragma unroll
    for (int i = 0; i < DD / 4; ++i) ar[i] = z;
}

__global__ void gnn_edge_max(const int* __restrict__ src, const int* __restrict__ dst,
                             const float* __restrict__ g, unsigned* __restrict__ menc, int e) {
    int i = blockIdx.x * blockDim.x + threadIdx.x;
    if (i >= e) return;
    atomicMax(&menc[dst[i]], fflip(g[src[i]]));
}

// decode max in place; seed denom with self-loop term exp(g - m)
__global__ void gnn_node_finalize(float* __restrict__ m, const float* __restrict__ g,
                                  float* __restrict__ denom, int n) {
    int v = blockIdx.x * blockDim.x + threadIdx.x;
    if (v >= n) return;
    float mv = funflip(((const unsigned*)m)[v]);
    m[v] = mv;
    denom[v] = expf(g[v] - mv);
}

__global__ void gnn_edge_denom(const int* __restrict__ src, const int* __restrict__ dst,
                               const float* __restrict__ g, const float* __restrict__ m,
                               float* __restrict__ denom, int e) {
    int i = blockIdx.x * blockDim.x + threadIdx.x;
    if (i >= e) return;
    int d = dst[i];
    atomicAdd(&denom[d], expf(g[src[i]] - m[d]));
}

// one wave per (edge or self-loop); lane covers 4 features
__global__ void gnn_edge_agg(const int* __restrict__ src, const int* __restrict__ dst,
                             const float* __restrict__ g, const float* __restrict__ m,
                             const float* __restrict__ denom, const float* __restrict__ h,
                             float* __restrict__ aggX, int e, int n) {
    int gid = blockIdx.x * blockDim.x + threadIdx.x;
    int edge = gid >> 5;
    int lane = gid & 31;
    if (edge >= e + n) return;
    int s, d;
    if (edge < e) { s = src[edge]; d = dst[edge]; }
    else          { s = d = edge - e; }
    float w = expf(g[s] - m[d]) / denom[d];
    float4 hv = *(const float4*)(h + (size_t)s * DD + lane * 4);
    float* out = aggX + (size_t)d * DD + lane * 4;
    atomicAdd(out + 0, w * hv.x);
    atomicAdd(out + 1, w * hv.y);
    atomicAdd(out + 2, w * hv.z);
    atomicAdd(out + 3, w * hv.w);
}

// ---------- fp32 WMMA GEMM: Out[n x 128] = hardswish(A[n x 128] @ W[128 x 128] + bias) ----
// 256 threads = 8 waves; wave w computes rows [blk*128 + w*16, +16) x all 128 cols.
// W staged into LDS in pair-interleaved fragment-ready layout:
//   pair-row p (= k>>1) holds (W[2p][c], W[2p+1][c]) contiguously -> one ds_load_b64 per B frag.
//   Pair-row stride 288 floats: 288 mod 64 = 32, so the two lane halves (pair-rows 2kk / 2kk+1)
//   hit disjoint 32-bank sets -> conflict-free.
// In-place safe (Out == A allowed): each wave only reads its own rows; stores after K-loop.
#define PRS 288   // floats per pair-row (128 pairs used, padded)
__launch_bounds__(256)
__global__ void gnn_gemm128_hswish(const float* __restrict__ A, const float* __restrict__ W,
                                   const float* __restrict__ bias, float* __restrict__ Out,
                                   int nrows) {
    __shared__ float sW[64 * PRS];   // 72 KB
    const int tid  = threadIdx.x;
    const int wave = tid >> 5;
    const int lane = tid & 31;
    const int row0 = blockIdx.x * 128;

    // stage W (128x128) into pair-interleaved LDS layout; global reads coalesced
    for (int i = tid; i < DD * DD; i += 256) {
        int k = i >> 7, col = i & 127;
        sW[(k >> 1) * PRS + col * 2 + (k & 1)] = W[i];
    }
    __syncthreads();

    const int mloc  = lane & 15;                // M (A rows) / N (B cols) within tile
    const int khalf = (lane < 16) ? 0 : 1;      // which K pair of the group of 4
    const int gr    = row0 + wave * 16 + mloc;  // global A/Out row for A-fragment
    const float* arow = A + (size_t)(gr < nrows ? gr : 0) * DD + khalf * 2;

    v8f acc[8];
#pragma unroll
    for (int t = 0; t < 8; ++t) acc[t] = (v8f){0.f, 0.f, 0.f, 0.f, 0.f, 0.f, 0.f, 0.f};

    for (int kk = 0; kk < DD / 4; ++kk) {
        float2 av = *(const float2*)(arow + kk * 4);
        v2f a; a.x = av.x; a.y = av.y;
        const float* wk = &sW[(kk * 2 + khalf) * PRS];
#pragma unroll
        for (int t = 0; t < 8; ++t) {
            v2f b = *(const v2f*)(wk + (t * 16 + mloc) * 2);
            acc[t] = __builtin_amdgcn_wmma_f32_16x16x4_f32(
                false, a, false, b, (short)0, acc[t], false, false);
        }
    }

    // D layout: VGPR r -> M = r (lanes 0-15) / r+8 (lanes 16-31); N = tile*16 + (lane&15)
    const int mbase = wave * 16 + ((lane < 16) ? 0 : 8);
    if (row0 + 128 <= nrows) {
        // full tile: unguarded stores
#pragma unroll
        for (int t = 0; t < 8; ++t) {
            int col = t * 16 + mloc;
            float bb = bias[col];
#pragma unroll
            for (int r = 0; r < 8; ++r) {
                Out[(size_t)(row0 + mbase + r) * DD + col] = hswish(acc[t][r] + bb);
            }
        }
    } else {
#pragma unroll
        for (int t = 0; t < 8; ++t) {
            int col = t * 16 + mloc;
            float bb = bias[col];
#pragma unroll
            for (int r = 0; r < 8; ++r) {
                int grow = row0 + mbase + r;
                if (grow < nrows) Out[(size_t)grow * DD + col] = hswish(acc[t][r] + bb);
            }
        }
    }
}

// ---------- head: logits (D x C, C<=64) + log_softmax ----------
__global__ void gnn_head(const float* __restrict__ h, const float* __restrict__ Wp2,
                         const float* __restrict__ bp2, float* __restrict__ out,
                         int n, int C) {
    int v = blockIdx.x * blockDim.x + threadIdx.x;
    if (v >= n) return;
    float logit[64];
    for (int c = 0; c < C; ++c) logit[c] = bp2[c];
    const float* hr = h + (size_t)v * DD;
    for (int k = 0; k < DD; ++k) {
        float hk = hr[k];
        const float* wrow = Wp2 + (size_t)k * C;
        for (int c = 0; c < C; ++c) logit[c] += hk * wrow[c];
    }
    float mx = logit[0];
    for (int c = 1; c < C; ++c) mx = fmaxf(mx, logit[c]);
    float ssum = 0.0f;
    for (int c = 0; c < C; ++c) ssum += expf(logit[c] - mx);
    float lse = mx + logf(ssum);
    for (int c = 0; c < C; ++c) out[(size_t)v * C + c] = logit[c] - lse;
}

extern "C" void kernel_launch(void* const* d_in, const int* in_sizes, int n_in,
                              void* d_out, int out_size, void* d_ws, size_t ws_size,
                              hipStream_t stream) {
    const float* x   = (const float*)d_in[0];
    const int*   src = (const int*)d_in[1];
    const int*   dst = (const int*)d_in[2];
    const float* Wg  = (const float*)d_in[3];
    const float* bg  = (const float*)d_in[4];
    const float* Wt  = (const float*)d_in[5];
    const float* bt  = (const float*)d_in[6];
    const float* W1  = (const float*)d_in[7];
    const float* b1  = (const float*)d_in[8];
    const float* bng = (const float*)d_in[9];
    const float* bnb = (const float*)d_in[10];
    const float* bnm = (const float*)d_in[11];
    const float* bnv = (const float*)d_in[12];
    const float* W2  = (const float*)d_in[13];
    const float* b2  = (const float*)d_in[14];
    const float* Wp1 = (const float*)d_in[15];
    const float* bp1 = (const float*)d_in[16];
    const float* Wp2 = (const float*)d_in[17];
    const float* bp2 = (const float*)d_in[18];

    const int n  = in_sizes[0] / DD;
    const int e  = in_sizes[1];
    const int Lt = in_sizes[4];      // bg is (L,1)
    const int C  = in_sizes[18];     // bp2 is (C,)

    // workspace layout (floats)
    float* ws = (float*)d_ws;
    size_t off = 0;
    float*    g     = ws + off; off += (size_t)n;
    float*    m     = ws + off; off += (size_t)n;          // unsigned during max phase
    float*    denom = ws + off; off += (size_t)n;
    float*    aggX  = ws + off; off += (size_t)n * DD;     // also reused as "u" (in-place gemm)
    float*    hbuf  = ws + off; off += (size_t)n * DD;
    float*    Wf    = ws + off; off += (size_t)Lt * DD * DD;
    float*    bf    = ws + off; off += (size_t)Lt * DD;
    (void)ws_size; (void)n_in; (void)out_size;

    const int nb = (n + 255) / 256;
    const int eb = (e + 255) / 256;

    gnn_fuse_weights<<<(Lt * DD * DD + 255) / 256, 256, 0, stream>>>(
        Wt, bt, W1, b1, bng, bnb, bnm, bnv, Wf, bf, Lt);

    const float* hin = x;
    for (int l = 0; l < Lt; ++l) {
        gnn_gate_init<<<nb, 256, 0, stream>>>(hin, Wg + l * DD, bg + l,
                                              g, (unsigned*)m, denom, aggX, n);
        gnn_edge_max<<<eb, 256, 0, stream>>>(src, dst, g, (unsigned*)m, e);
        gnn_node_finalize<<<nb, 256, 0, stream>>>(m, g, denom, n);
        gnn_edge_denom<<<eb, 256, 0, stream>>>(src, dst, g, m, denom, e);
        const int en = e + n;
        gnn_edge_agg<<<(int)(((long long)en * 32 + 255) / 256), 256, 0, stream>>>(
            src, dst, g, m, denom, hin, aggX, e, n);
        // fused (Wt@W1 * bn_scale) transform, in place on aggX
        gnn_gemm128_hswish<<<(n + 127) / 128, 256, 0, stream>>>(
            aggX, Wf + (size_t)l * DD * DD, bf + (size_t)l * DD, aggX, n);
        // second linear + outer hardswish -> hbuf (safe: reads only aggX)
        gnn_gemm128_hswish<<<(n + 127) / 128, 256, 0, stream>>>(
            aggX, W2 + (size_t)l * DD * DD, b2 + (size_t)l * DD, hbuf, n);
        hin = hbuf;
    }

    // post-MP head: in-place gemm on hbuf, then logits + log_softmax
    gnn_gemm128_hswish<<<(n + 127) / 128, 256, 0, stream>>>(hbuf, Wp1, bp1, hbuf, n);
    gnn_head<<<nb, 256, 0, stream>>>(hbuf, Wp2, bp2, (float*)d_out, n, C);
}